// ExponentialMovingAverage_23682449670505
// MI455X (gfx1250) — compile-verified
//
#include <hip/hip_runtime.h>

typedef float v2f __attribute__((ext_vector_type(2)));
typedef float v8f __attribute__((ext_vector_type(8)));

namespace {
constexpr int Bn = 64, Tn = 4096, Dn = 256;
constexpr float S = 0.3f;
constexpr float R = 1.0f - S;        // 0.7
constexpr int TC   = 256;            // chunk length (timesteps per wave)
constexpr int HALO = 64;             // r^64 ~ 1.2e-10: carry rebuild exact to f32 eps
constexpr int NCHG   = Dn / 16;      // 16 channel groups
constexpr int NCHUNK = Tn / TC;      // 16 chunks
constexpr int WAVES  = Bn * NCHUNK * NCHG;  // 16384 waves
constexpr int WPB    = 8;            // waves per block (256 threads)

constexpr float cpow(float b, int e) { float r = 1.f; for (int i = 0; i < e; ++i) r *= b; return r; }
}

__device__ inline v8f wmma_f32_16x16x4(v2f a, v2f b, v8f c) {
  // D(16x16,f32) = A(16x4,f32) * B(4x16,f32) + C
  return __builtin_amdgcn_wmma_f32_16x16x4_f32(false, a, false, b, (short)0, c, false, false);
}

__global__ __launch_bounds__(WPB * 32)
void ema_scan_wmma(const float* __restrict__ x, float* __restrict__ y) {
  const int lane = threadIdx.x & 31;
  const int n    = lane & 15;   // matrix N column (channel) and A's M row
  const int half = lane >> 4;   // selects K pair for A/B striping

  const int wid   = blockIdx.x * WPB + (threadIdx.x >> 5);
  const int chg   = wid & (NCHG - 1);
  const int chunk = (wid >> 4) & (NCHUNK - 1);
  const int b     = wid >> 8;

  // r^n for n = 0..15 via square-and-multiply (pure selects, no divergence)
  float rn = 1.f;
  rn *= (n & 1) ? R : 1.f;
  rn *= (n & 2) ? cpow(R, 2) : 1.f;
  rn *= (n & 4) ? cpow(R, 4) : 1.f;
  rn *= (n & 8) ? cpow(R, 8) : 1.f;

  constexpr float INV = 1.0f / R;
  const float hinv = half ? (INV * INV) : 1.f;

  // A tiles (constants): a[q][v] = M[n][j], j = 4q + 2*half + v,
  // M[i][j] = (i>=j) ? s*r^(i-j) : 0   (lower-triangular scan matrix)
  v2f a[4];
#pragma unroll
  for (int q = 0; q < 4; ++q) {
#pragma unroll
    for (int v = 0; v < 2; ++v) {
      const int j = 4 * q + 2 * half + v;
      const float coef = S * rn * cpow(INV, 4 * q + v) * hinv;   // s * r^(n-j)
      a[q][v] = (n >= j) ? coef : 0.f;
    }
  }

  // carry weights rc[v] = r^(i+1), i = v + 8*half  (rank-1 carry into C init)
  float rc[8];
  const float hr = half ? cpow(R, 8) : 1.f;
#pragma unroll
  for (int v = 0; v < 8; ++v) rc[v] = cpow(R, v + 1) * hr;

  const int nhalo  = (chunk == 0) ? 0 : (HALO / 16);
  const int nblk   = nhalo + TC / 16;
  const int tfirst = chunk * TC - nhalo * 16;
  const int ch0    = chg * 16;

  const float* xb = x + ((size_t)b * Tn + tfirst) * (size_t)Dn + ch0 + n;
  float*       yb = y + ((size_t)b * Tn) * (size_t)Dn + ch0 + n;

  // chunk 0: virtual carry x[0] makes y[0] = (s+r)*x[0] = x[0]
  float yprev = (chunk == 0) ? xb[0] : 0.f;

  // B tile for block 0: bt[q][v] = x[t = tfirst + 2*half + 4q + v][ch0+n]
  const float* px = xb + (size_t)(2 * half) * Dn;
  v2f bt[4];
#pragma unroll
  for (int q = 0; q < 4; ++q) {
    bt[q][0] = px[(size_t)(4 * q + 0) * Dn];
    bt[q][1] = px[(size_t)(4 * q + 1) * Dn];
  }

  for (int blk = 0; blk < nblk; ++blk) {
    // prefetch next block's B tiles so loads overlap the WMMA chain
    const float* pn = px + (size_t)16 * Dn;
    v2f bnx[4] = {bt[0], bt[1], bt[2], bt[3]};
    if (blk + 1 < nblk) {
#pragma unroll
      for (int q = 0; q < 4; ++q) {
        bnx[q][0] = pn[(size_t)(4 * q + 0) * Dn];
        bnx[q][1] = pn[(size_t)(4 * q + 1) * Dn];
      }
    }

    // C init = r^(i+1) * y_prev  (carry from previous 16-block)
    v8f c;
#pragma unroll
    for (int v = 0; v < 8; ++v) c[v] = rc[v] * yprev;

    // Y(16x16) = M(16x16) * X(16x16) + C, chained over K in steps of 4
    c = wmma_f32_16x16x4(a[0], bt[0], c);
    c = wmma_f32_16x16x4(a[1], bt[1], c);
    c = wmma_f32_16x16x4(a[2], bt[2], c);
    c = wmma_f32_16x16x4(a[3], bt[3], c);

    if (blk >= nhalo) {   // wave-uniform: halo blocks only rebuild the carry
      const int tb = tfirst + 16 * blk;
      float* ps = yb + ((size_t)tb + 8 * half) * (size_t)Dn;
#pragma unroll
      for (int v = 0; v < 8; ++v)
        __builtin_nontemporal_store(c[v], ps + (size_t)v * Dn);   // y is write-once
    }

    // y_prev[n] = Y[15][n], held in d[7] of lane 16+n
    yprev = __shfl(c[7], 16 + n, 32);

    px = pn;
#pragma unroll
    for (int q = 0; q < 4; ++q) bt[q] = bnx[q];
  }
}

extern "C" void kernel_launch(void* const* d_in, const int* in_sizes, int n_in,
                              void* d_out, int out_size, void* d_ws, size_t ws_size,
                              hipStream_t stream) {
  (void)in_sizes; (void)n_in; (void)d_ws; (void)ws_size; (void)out_size;
  const float* x = (const float*)d_in[0];
  float* y = (float*)d_out;
  dim3 grid(WAVES / WPB), block(WPB * 32);
  hipLaunchKernelGGL(ema_scan_wmma, grid, block, 0, stream, x, y);
}